// VisionMamba_patch_RoPE_2302102471249
// MI455X (gfx1250) — compile-verified
//
#include <hip/hip_runtime.h>
#include <math.h>

// ---------------- problem constants ----------------
#define BATCH   2
#define TT      7
#define HP      14
#define WPP     14
#define NP      (HP*WPP)        // 196
#define LTOT    (TT*NP)         // 1372
#define MTOK    (BATCH*LTOT)    // 2744
#define CDIM    192
#define DN      384
#define NSTATE  16
#define RRANK   12
#define DEPTH   4
#define KPATCH  768             // 3*16*16
#define XDBLW   (RRANK+2*NSTATE) // 44
#define IMGHW   224

typedef __attribute__((ext_vector_type(2))) float v2f;
typedef __attribute__((ext_vector_type(8))) float v8f;

// ============================================================
// Generic WMMA f32 GEMM:  C[M,N] = A[M,K] * W[N,K]^T  (+ epilogue)
// One wave computes a 16x16 output tile via V_WMMA_F32_16X16X4_F32.
// A-layout per ISA: lane l holds row M=l%16, a={A[M][2*hi],A[M][2*hi+1]}, hi=l/16.
// B (K x N) mirrored: lane l holds col N=l%16, b={B[2*hi][N],B[2*hi+1][N]} = W[N][2*hi..].
// OOB rows/cols are CLAMPED (not predicated): A-row m only affects D-row m and
// B-col n only affects D-col n, and the store guard drops those — so clamped
// loads are harmless and the inner loop stays divergence-free (EXEC all-1s,
// as WMMA requires).
// ep: 0 = plain, 1 = +bias[col], 2 = softplus(x + bias[col])
// ============================================================
__global__ __launch_bounds__(256) void vm_gemm_wmma(
    const float* __restrict__ A, int lda,
    const float* __restrict__ W, int ldw,
    float* __restrict__ Cout, int ldc,
    int M, int N, int K,
    const float* __restrict__ bias, int ep)
{
    const int wave = threadIdx.x >> 5;
    const int lane = threadIdx.x & 31;
    const int ntn  = (N + 15) >> 4;
    const int ntm  = (M + 15) >> 4;
    const int tile = blockIdx.x * (blockDim.x >> 5) + wave;
    if (tile >= ntm * ntn) return;          // wave-uniform exit (EXEC stays full)
    const int tm = tile / ntn, tn = tile % ntn;
    const int m0 = tm << 4,   n0 = tn << 4;
    const int r  = lane & 15, hi = lane >> 4;

    int am = m0 + r; if (am > M - 1) am = M - 1;   // clamp: divergence-free loads
    int wn = n0 + r; if (wn > N - 1) wn = N - 1;
    const float* ap = A + (size_t)am * lda + 2*hi;
    const float* wp = W + (size_t)wn * ldw + 2*hi;

    v8f acc = {0.f,0.f,0.f,0.f,0.f,0.f,0.f,0.f};
    for (int k0 = 0; k0 < K; k0 += 4) {
        if ((k0 & 63) == 0 && k0 + 64 < K) {   // CDNA5 prefetch path
            __builtin_prefetch(ap + k0 + 64, 0, 3);
            __builtin_prefetch(wp + k0 + 64, 0, 3);
        }
        v2f a = *(const v2f*)(ap + k0);
        v2f b = *(const v2f*)(wp + k0);
        acc = __builtin_amdgcn_wmma_f32_16x16x4_f32(
                  false, a, false, b, (short)0, acc, false, false);
    }

    const int col = n0 + r;
    if (col < N) {
        float bv = (ep >= 1) ? bias[col] : 0.0f;
        #pragma unroll
        for (int rr = 0; rr < 8; rr++) {
            int row = m0 + hi*8 + rr;       // C/D layout: lanes16-31 hold rows 8..15
            if (row < M) {
                float v = acc[rr] + bv;
                if (ep == 2) v = (v > 20.0f) ? v : log1pf(__expf(v));
                Cout[(size_t)row * ldc + col] = v;
            }
        }
    }
}

// ---------------- im2row of input video into (MTOK, 768) ----------------
__global__ void vm_im2row(const float* __restrict__ x, float* __restrict__ A0)
{
    int idx = blockIdx.x * 256 + threadIdx.x;
    if (idx >= MTOK * KPATCH) return;
    int m = idx / KPATCH, k = idx % KPATCH;
    int b = m / LTOT, rm = m % LTOT;
    int t = rm / NP,  hw = rm % NP;
    int hp = hw / WPP, wp = hw % WPP;
    int c = k >> 8, pq = k & 255, p = pq >> 4, q = pq & 15;
    A0[idx] = x[ (((size_t)(b*3 + c)*TT + t)*IMGHW + hp*16 + p) * IMGHW + wp*16 + q ];
}

// ---------------- add tpos (patch GEMM epilogue already added pb) ----------------
__global__ void vm_add_tpos(float* __restrict__ h, const float* __restrict__ tpos)
{
    int idx = blockIdx.x * 256 + threadIdx.x;
    if (idx >= MTOK * CDIM) return;
    int m = idx / CDIM, d = idx % CDIM;
    int t = (m % LTOT) / NP;
    h[idx] += tpos[t * CDIM + d];
}

// ---------------- RoPE cos/sin tables (LTOT x C) ----------------
__global__ void vm_rope_table(float* __restrict__ cosT, float* __restrict__ sinT)
{
    int idx = blockIdx.x * 256 + threadIdx.x;
    if (idx >= LTOT * CDIM) return;
    int pos = idx / CDIM, d = idx % CDIM;
    int hw = pos % NP;
    int hp = hw / WPP, wp = hw % WPP;
    const float PI = 3.14159265358979323846f;
    float val;
    if (d < 96) {
        int j = d >> 1;
        float freq = (1.0f + 4.0f * (float)j / 47.0f) * PI;   // linspace(1,5,48)*pi
        float th = -1.0f + 2.0f * (float)hp / 13.0f;          // linspace(-1,1,14)
        val = th * freq;
    } else {
        int j = (d - 96) >> 1;
        float freq = (1.0f + 4.0f * (float)j / 47.0f) * PI;
        float tw = -1.0f + 2.0f * (float)wp / 13.0f;
        val = tw * freq;
    }
    cosT[idx] = cosf(val);
    sinT[idx] = sinf(val);
}

// ---------------- per-layer: rope(h) [+ rope(res)] -> res; rmsnorm -> hn ----------------
__global__ __launch_bounds__(192) void vm_layerpre(
    const float* __restrict__ h, float* __restrict__ res, float* __restrict__ hn,
    const float* __restrict__ cosT, const float* __restrict__ sinT,
    const float* __restrict__ lnw, int first)
{
    __shared__ float sred[256];
    __shared__ float shr[CDIM];
    int tok = blockIdx.x;
    int d = threadIdx.x;
    int pos = tok % LTOT;
    const float* cp = cosT + (size_t)pos * CDIM;
    const float* sp = sinT + (size_t)pos * CDIM;
    int dp = d ^ 1;
    float sgn = (d & 1) ? 1.0f : -1.0f;

    float hv = h[(size_t)tok*CDIM + d];
    float hq = h[(size_t)tok*CDIM + dp];
    float rh = hv * cp[d] + sgn * hq * sp[d];

    float rv;
    if (first) {
        rv = rh;
    } else {
        shr[d] = res[(size_t)tok*CDIM + d];
        __syncthreads();
        float rr0 = shr[d], rr1 = shr[dp];
        rv = rh + rr0 * cp[d] + sgn * rr1 * sp[d];
    }
    res[(size_t)tok*CDIM + d] = rv;

    sred[d] = rv * rv;
    if (d < 64) sred[192 + d] = 0.0f;
    __syncthreads();
    for (int s = 128; s > 0; s >>= 1) {
        if (d < s) sred[d] += sred[d + s];
        __syncthreads();
    }
    float inv = rsqrtf(sred[0] / (float)CDIM + 1e-5f);
    hn[(size_t)tok*CDIM + d] = rv * inv * lnw[d];
}

// ---------------- final: res += h; hn = rmsnorm(res)*fnw ----------------
__global__ __launch_bounds__(192) void vm_finalnorm(
    const float* __restrict__ h, float* __restrict__ res, float* __restrict__ hn,
    const float* __restrict__ fnw)
{
    __shared__ float sred[256];
    int tok = blockIdx.x;
    int d = threadIdx.x;
    float rv = h[(size_t)tok*CDIM + d] + res[(size_t)tok*CDIM + d];
    sred[d] = rv * rv;
    if (d < 64) sred[192 + d] = 0.0f;
    __syncthreads();
    for (int s = 128; s > 0; s >>= 1) {
        if (d < s) sred[d] += sred[d + s];
        __syncthreads();
    }
    float inv = rsqrtf(sred[0] / (float)CDIM + 1e-5f);
    hn[(size_t)tok*CDIM + d] = rv * inv * fnw[d];
}

// ---------------- causal depthwise conv (K=4) + SiLU; rev maps reversed sequence ----------------
__global__ void vm_dwconv_silu(
    const float* __restrict__ xz, const float* __restrict__ cw,
    const float* __restrict__ cb, float* __restrict__ xc, int rev)
{
    int idx = blockIdx.x * 256 + threadIdx.x;
    if (idx >= BATCH * LTOT * DN) return;
    int d = idx % DN;
    int rem = idx / DN;
    int t = rem % LTOT;
    int b = rem / LTOT;
    float acc = cb[d];
    #pragma unroll
    for (int k = 0; k < 4; k++) {
        int s = t - 3 + k;
        if (s >= 0) {
            int si = rev ? (LTOT - 1 - s) : s;
            acc += cw[d*4 + k] * xz[((size_t)(b*LTOT + si)) * (2*DN) + d];
        }
    }
    acc = acc / (1.0f + __expf(-acc));   // silu
    xc[((size_t)(b*LTOT + t)) * DN + d] = acc;
}

// ---------------- selective scan: 1 thread per channel, h[16] in registers ----------------
__global__ __launch_bounds__(384) void vm_scan(
    const float* __restrict__ xc, const float* __restrict__ dtv,
    const float* __restrict__ xdbl, const float* __restrict__ Alog,
    const float* __restrict__ Dp, float* __restrict__ y, int rev)
{
    int b = blockIdx.x;
    int d = threadIdx.x;
    float Av[NSTATE];
    #pragma unroll
    for (int n = 0; n < NSTATE; n++) Av[n] = -__expf(Alog[d*NSTATE + n]);
    float hreg[NSTATE];
    #pragma unroll
    for (int n = 0; n < NSTATE; n++) hreg[n] = 0.0f;
    float Dd = Dp[d];

    for (int t = 0; t < LTOT; t++) {
        size_t row = (size_t)(b*LTOT + t);
        const float* xr = xdbl + row * XDBLW;      // [0:12)=dtproj-in, [12:28)=B, [28:44)=C
        float dt = dtv[row * DN + d];
        float u  = xc [row * DN + d];
        float du = dt * u;
        float acc = 0.0f;
        #pragma unroll
        for (int n = 0; n < NSTATE; n++) {
            float Bn = xr[RRANK + n];
            float Cn = xr[RRANK + NSTATE + n];
            hreg[n] = __expf(dt * Av[n]) * hreg[n] + du * Bn;
            acc += hreg[n] * Cn;
        }
        int pos = rev ? (LTOT - 1 - t) : t;
        y[((size_t)(b*LTOT + pos)) * DN + d] = acc + u * Dd;
    }
}

// ---------------- gate: yc = (yf + yb) * silu(z) ----------------
__global__ void vm_gate(const float* __restrict__ xz, const float* __restrict__ yf,
                        const float* __restrict__ yb, float* __restrict__ yc)
{
    int idx = blockIdx.x * 256 + threadIdx.x;
    if (idx >= BATCH * LTOT * DN) return;
    int d = idx % DN;
    int row = idx / DN;
    float z = xz[(size_t)row * (2*DN) + DN + d];
    float s = z / (1.0f + __expf(-z));
    yc[idx] = (yf[idx] + yb[idx]) * s;
}

// ---------------- transpose uw (192 x 768, d-major) -> uwT (768 x 192) ----------------
__global__ void vm_transpose_uw(const float* __restrict__ uw, float* __restrict__ uwT)
{
    int idx = blockIdx.x * 256 + threadIdx.x;
    if (idx >= CDIM * KPATCH) return;
    int d = idx / KPATCH, n = idx % KPATCH;
    uwT[(size_t)n * CDIM + d] = uw[idx];
}

// ---------------- scatter un-patch GEMM result into output + ub ----------------
__global__ void vm_scatter(const float* __restrict__ up, const float* __restrict__ ub,
                           float* __restrict__ out)
{
    int idx = blockIdx.x * 256 + threadIdx.x;
    if (idx >= MTOK * KPATCH) return;
    int m = idx / KPATCH, n = idx % KPATCH;
    int b = m / LTOT, rm = m % LTOT;
    int t = rm / NP,  hw = rm % NP;
    int hp = hw / WPP, wp = hw % WPP;
    int c = n >> 8, pq = n & 255, p = pq >> 4, q = pq & 15;
    out[ (((size_t)(b*3 + c)*TT + t)*IMGHW + hp*16 + p) * IMGHW + wp*16 + q ]
        = up[idx] + ub[c];
}

// ============================================================
// host-side launch
// ============================================================
static inline int ceil_div(int a, int b) { return (a + b - 1) / b; }

static void launch_gemm(const float* A, int lda, const float* W, int ldw,
                        float* C, int ldc, int M, int N, int K,
                        const float* bias, int ep, hipStream_t stream)
{
    int tiles = ceil_div(M,16) * ceil_div(N,16);
    int blocks = ceil_div(tiles, 8);
    vm_gemm_wmma<<<blocks, 256, 0, stream>>>(A, lda, W, ldw, C, ldc, M, N, K, bias, ep);
}

extern "C" void kernel_launch(void* const* d_in, const int* in_sizes, int n_in,
                              void* d_out, int out_size, void* d_ws, size_t ws_size,
                              hipStream_t stream)
{
    const float* x      = (const float*)d_in[0];
    const float* pw     = (const float*)d_in[1];
    const float* pb     = (const float*)d_in[2];
    const float* tpos   = (const float*)d_in[3];
    const float* ln_w   = (const float*)d_in[4];
    const float* ipw    = (const float*)d_in[5];
    const float* cw_f   = (const float*)d_in[6];
    const float* cb_f   = (const float*)d_in[7];
    const float* xpw_f  = (const float*)d_in[8];
    const float* dtw_f  = (const float*)d_in[9];
    const float* dtb_f  = (const float*)d_in[10];
    const float* Alog_f = (const float*)d_in[11];
    const float* Dp_f   = (const float*)d_in[12];
    const float* cw_b   = (const float*)d_in[13];
    const float* cb_b   = (const float*)d_in[14];
    const float* xpw_b  = (const float*)d_in[15];
    const float* dtw_b  = (const float*)d_in[16];
    const float* dtb_b  = (const float*)d_in[17];
    const float* Alog_b = (const float*)d_in[18];
    const float* Dp_b   = (const float*)d_in[19];
    const float* opw    = (const float*)d_in[20];
    const float* fnw    = (const float*)d_in[21];
    const float* uw     = (const float*)d_in[22];
    const float* ub     = (const float*)d_in[23];
    float* out = (float*)d_out;

    // workspace layout (~65 MB of f32; all sizes even -> 8B alignment for v2f loads)
    float* wsf = (float*)d_ws;
    size_t off = 0;
    auto alloc = [&](size_t n) { float* p = wsf + off; off += n; return p; };
    float* A0   = alloc((size_t)MTOK * KPATCH);
    float* cosT = alloc((size_t)LTOT * CDIM);
    float* sinT = alloc((size_t)LTOT * CDIM);
    float* hbuf = alloc((size_t)MTOK * CDIM);
    float* res  = alloc((size_t)MTOK * CDIM);
    float* hn   = alloc((size_t)MTOK * CDIM);
    float* xz   = alloc((size_t)MTOK * 2 * DN);
    float* xcf  = alloc((size_t)MTOK * DN);
    float* xcb  = alloc((size_t)MTOK * DN);
    float* xdf  = alloc((size_t)MTOK * XDBLW);
    float* xdb  = alloc((size_t)MTOK * XDBLW);
    float* dtfb = alloc((size_t)MTOK * DN);
    float* dtbb = alloc((size_t)MTOK * DN);
    float* yf   = alloc((size_t)MTOK * DN);
    float* yb   = alloc((size_t)MTOK * DN);
    float* yc   = alloc((size_t)MTOK * DN);
    float* uwT  = alloc((size_t)KPATCH * CDIM);
    float* up   = alloc((size_t)MTOK * KPATCH);
    (void)ws_size; // requires ~65MB; harness-provided ws assumed sufficient

    // ---- patch embedding ----
    vm_im2row<<<ceil_div(MTOK*KPATCH,256), 256, 0, stream>>>(x, A0);
    vm_rope_table<<<ceil_div(LTOT*CDIM,256), 256, 0, stream>>>(cosT, sinT);
    launch_gemm(A0, KPATCH, pw, KPATCH, hbuf, CDIM, MTOK, CDIM, KPATCH, pb, 1, stream);
    vm_add_tpos<<<ceil_div(MTOK*CDIM,256), 256, 0, stream>>>(hbuf, tpos);

    // ---- layers ----
    for (int i = 0; i < DEPTH; i++) {
        const float* ipw_i  = ipw    + (size_t)i * (2*DN) * CDIM;
        const float* lnw_i  = ln_w   + (size_t)i * CDIM;
        const float* cwf_i  = cw_f   + (size_t)i * DN * 4;
        const float* cbf_i  = cb_f   + (size_t)i * DN;
        const float* xpwf_i = xpw_f  + (size_t)i * XDBLW * DN;
        const float* dtwf_i = dtw_f  + (size_t)i * DN * RRANK;
        const float* dtbf_i = dtb_f  + (size_t)i * DN;
        const float* Alf_i  = Alog_f + (size_t)i * DN * NSTATE;
        const float* Dpf_i  = Dp_f   + (size_t)i * DN;
        const float* cwb_i  = cw_b   + (size_t)i * DN * 4;
        const float* cbb_i  = cb_b   + (size_t)i * DN;
        const float* xpwb_i = xpw_b  + (size_t)i * XDBLW * DN;
        const float* dtwb_i = dtw_b  + (size_t)i * DN * RRANK;
        const float* dtbb_i = dtb_b  + (size_t)i * DN;
        const float* Alb_i  = Alog_b + (size_t)i * DN * NSTATE;
        const float* Dpb_i  = Dp_b   + (size_t)i * DN;
        const float* opw_i  = opw    + (size_t)i * CDIM * DN;

        vm_layerpre<<<MTOK, 192, 0, stream>>>(hbuf, res, hn, cosT, sinT, lnw_i, i == 0);

        // xz = hn @ ipw^T   (2744 x 768)
        launch_gemm(hn, CDIM, ipw_i, CDIM, xz, 2*DN, MTOK, 2*DN, CDIM, nullptr, 0, stream);

        // forward branch
        vm_dwconv_silu<<<ceil_div(BATCH*LTOT*DN,256), 256, 0, stream>>>(xz, cwf_i, cbf_i, xcf, 0);
        launch_gemm(xcf, DN, xpwf_i, DN, xdf, XDBLW, MTOK, XDBLW, DN, nullptr, 0, stream);
        launch_gemm(xdf, XDBLW, dtwf_i, RRANK, dtfb, DN, MTOK, DN, RRANK, dtbf_i, 2, stream);
        vm_scan<<<BATCH, DN, 0, stream>>>(xcf, dtfb, xdf, Alf_i, Dpf_i, yf, 0);

        // backward branch (operates on reversed sequence, writes at original positions)
        vm_dwconv_silu<<<ceil_div(BATCH*LTOT*DN,256), 256, 0, stream>>>(xz, cwb_i, cbb_i, xcb, 1);
        launch_gemm(xcb, DN, xpwb_i, DN, xdb, XDBLW, MTOK, XDBLW, DN, nullptr, 0, stream);
        launch_gemm(xdb, XDBLW, dtwb_i, RRANK, dtbb, DN, MTOK, DN, RRANK, dtbb_i, 2, stream);
        vm_scan<<<BATCH, DN, 0, stream>>>(xcb, dtbb, xdb, Alb_i, Dpb_i, yb, 1);

        // gate + output projection -> h
        vm_gate<<<ceil_div(BATCH*LTOT*DN,256), 256, 0, stream>>>(xz, yf, yb, yc);
        launch_gemm(yc, DN, opw_i, DN, hbuf, CDIM, MTOK, CDIM, DN, nullptr, 0, stream);
    }

    // ---- final norm + un-patch ----
    vm_finalnorm<<<MTOK, 192, 0, stream>>>(hbuf, res, hn, fnw);
    vm_transpose_uw<<<ceil_div(CDIM*KPATCH,256), 256, 0, stream>>>(uw, uwT);
    launch_gemm(hn, CDIM, uwT, CDIM, up, KPATCH, MTOK, KPATCH, CDIM, nullptr, 0, stream);
    vm_scatter<<<ceil_div(MTOK*KPATCH,256), 256, 0, stream>>>(up, ub, out);
    (void)in_sizes; (void)n_in; (void)out_size;
}